// VectorQuantize_23545010717323
// MI455X (gfx1250) — compile-verified
//
#include <hip/hip_runtime.h>
#include <cstdint>

typedef __attribute__((ext_vector_type(2))) float v2f;
typedef __attribute__((ext_vector_type(8))) float v8f;
typedef __attribute__((ext_vector_type(4))) unsigned int u32x4;
typedef __attribute__((ext_vector_type(8))) int i32x8;
typedef __attribute__((ext_vector_type(4))) int i32x4;

#define N_ROWS 32768
#define DDIM   256
#define KCODES 8192
#define TILEW  32                    // codes per TDM tile
#define TILES  (KCODES / TILEW)      // 256
#define ROWS_PER_WG 128              // 8 waves * 16 rows
#define LDSROW 260                   // 256 + 4 pad floats (bank-conflict-free)
#define TILE_F (TILEW * LDSROW)      // 8320 floats = 33280 B per buffer

// ---------------------------------------------------------------------------
// TDM: load a 32-code tile from transposed embeddings embT[K=8192][D=256]
// into LDS as 32 rows of 256 floats, padded to 260 floats per row.
//   tensor: dim0 = D = 256 (contiguous), dim1 = K = 8192, stride0 = 256
//   tile:   dim0 = 256, dim1 = 32
//   pad:    every 256 DWORDs (code 7) add 4 DWORDs (code 3)
// ---------------------------------------------------------------------------
__device__ __forceinline__ void tdm_load_tile(const float* embT, int kt,
                                              unsigned lds_byte_off) {
  unsigned long long ga =
      (unsigned long long)(uintptr_t)(embT + (size_t)kt * TILEW * DDIM);
  u32x4 g0;
  g0.x = 1u;                                            // count=1, user mode
  g0.y = lds_byte_off;                                  // lds_addr [63:32]
  g0.z = (unsigned)(ga & 0xffffffffu);                  // global_addr[31:0]
  g0.w = ((unsigned)(ga >> 32) & 0x1ffffffu) | (2u << 30); // ga[56:32], type=2
  i32x8 g1;
  // data_size=2 (4B) | pad_enable | pad_interval=7 (256 DW) | pad_amount=3 (4 DW)
  g1[0] = (int)((2u << 16) | (1u << 20) | (7u << 22) | (3u << 25));
  g1[1] = (int)(0x100u << 16);      // tensor_dim0 = 256 (low 16)
  g1[2] = (int)(0x2000u << 16);     // tensor_dim0 hi=0 ; tensor_dim1 = 8192
  g1[3] = (int)(0x100u << 16);      // tensor_dim1 hi=0 ; tile_dim0 = 256
  g1[4] = TILEW;                    // tile_dim1 = 32 ; tile_dim2 = 0
  g1[5] = 256;                      // tensor_dim0_stride = 256
  g1[6] = 0;                        // stride0 hi ; tensor_dim1_stride lo
  g1[7] = 0;
  i32x4 z4 = {0, 0, 0, 0};
  i32x8 z8 = {0, 0, 0, 0, 0, 0, 0, 0};
  __builtin_amdgcn_tensor_load_to_lds(g0, g1, z4, z4, z8, 0);
}

// ---------------------------------------------------------------------------
// Transpose emb [D=256][K=8192] -> embT [K=8192][D=256] (tiled via LDS)
// grid (K/32, D/32), block 256 = 32x8
// ---------------------------------------------------------------------------
__global__ void vq_transpose_kernel(const float* __restrict__ emb,
                                    float* __restrict__ embT) {
  __shared__ float t[32][33];
  const int tx = threadIdx.x & 31;
  const int ty = threadIdx.x >> 5;
  const int k0 = blockIdx.x * 32;
  const int d0 = blockIdx.y * 32;
#pragma unroll
  for (int i = 0; i < 32; i += 8)
    t[ty + i][tx] = emb[(size_t)(d0 + ty + i) * KCODES + k0 + tx];
  __syncthreads();
#pragma unroll
  for (int i = 0; i < 32; i += 8)
    embT[(size_t)(k0 + ty + i) * DDIM + d0 + tx] = t[tx][ty + i];
}

// ---------------------------------------------------------------------------
// Kernel A: enorm[k] = sum_d emb[d][k]^2
// ---------------------------------------------------------------------------
__global__ void vq_enorm_kernel(const float* __restrict__ emb,
                                float* __restrict__ enorm) {
  int k = blockIdx.x * 256 + threadIdx.x;
  float s = 0.f;
  for (int d = 0; d < DDIM; ++d) {
    float v = emb[(size_t)d * KCODES + k];
    s += v * v;
  }
  enorm[k] = s;
}

__global__ void vq_init_loss_kernel(float* loss) { *loss = 0.f; }

// ---------------------------------------------------------------------------
// Kernel B: fused GEMM (f32 WMMA) + argmin over K per row.
// grid = 256 workgroups (128 rows each), block = 256 threads (8 waves).
// Each wave: 16 rows, A resident in 128 VGPRs, two interleaved WMMA chains
// over a 32-code LDS tile fed by the Tensor Data Mover (double-buffered).
// ---------------------------------------------------------------------------
__global__ void __launch_bounds__(256)
vq_argmin_kernel(const float* __restrict__ x, const float* __restrict__ embT,
                 const float* __restrict__ enormG, int* __restrict__ idxOut) {
  __shared__ float btile[2 * TILE_F];      // double-buffered tile (66560 B)

  const int tid   = threadIdx.x;
  const int lane  = tid & 31;
  const int wave  = tid >> 5;
  const int m     = lane & 15;             // column within 16-code subtile
  const int khalf = (lane >> 4) << 1;      // 0 or 2 (K-pair per lane half)
  const int row0  = blockIdx.x * ROWS_PER_WG + wave * 16;

  const unsigned lds_base = (unsigned)(uintptr_t)(void*)&btile[0];

  // Prime the pipeline: wave 0 DMAs tile 0 into buffer 0.
  if (wave == 0) tdm_load_tile(embT, 0, lds_base);

  // A fragments for this wave's 16 rows: a[s] covers D columns 4s..4s+3.
  // WMMA f32 16x16x4 A layout: lane<16 -> (M=lane, K=0/1), lane>=16 -> (M=lane-16, K=2/3)
  v2f a[64];
  {
    const float* arow = x + (size_t)(row0 + m) * DDIM + khalf;
#pragma unroll
    for (int s = 0; s < 64; ++s) a[s] = *(const v2f*)(arow + 4 * s);
  }

  float best[8];
  int   bidx[8];
#pragma unroll
  for (int j = 0; j < 8; ++j) { best[j] = 3.4e38f; bidx[j] = 0; }

  // Buffer bases toggled per tile (avoids per-iteration select math).
  const float* Bbase0 = btile + m * LDSROW + khalf;         // codes 0..15
  const float* Bbase1 = btile + (m + 16) * LDSROW + khalf;  // codes 16..31

  for (int t = 0; t < TILES; ++t) {
    if (wave == 0) __builtin_amdgcn_s_wait_tensorcnt(0);  // tile t landed
    __syncthreads();                                      // visible to all waves
    if (wave == 0 && t + 1 < TILES)                       // prefetch next tile
      tdm_load_tile(embT, t + 1, lds_base + ((t + 1) & 1) * (TILE_F * 4));

    // Hoist the enorm loads so global latency overlaps the compute loop.
    const float en0 = enormG[t * TILEW + m];
    const float en1 = enormG[t * TILEW + 16 + m];

    // Each lane's B fragment (d, d+1) is contiguous -> aligned ds_load_b64
    // with immediate offsets 16*s bytes. Bank-conflict-free (bank 4m mod 64).
    const int buf = (t & 1) * TILE_F;
    const float* Brow0 = Bbase0 + buf;
    const float* Brow1 = Bbase1 + buf;

    v8f c0 = {}, c1 = {};
    // Software pipeline, prefetch distance 2: ring of two fragment pairs.
    v2f p0[2], p1[2];
    p0[0] = *(const v2f*)(Brow0);     p1[0] = *(const v2f*)(Brow1);
    p0[1] = *(const v2f*)(Brow0 + 4); p1[1] = *(const v2f*)(Brow1 + 4);
#pragma unroll
    for (int s = 0; s < 64; ++s) {
      v2f b0 = p0[s & 1], b1 = p1[s & 1];
      if (s < 62) {
        p0[s & 1] = *(const v2f*)(Brow0 + 4 * (s + 2));
        p1[s & 1] = *(const v2f*)(Brow1 + 4 * (s + 2));
      }
      c0 = __builtin_amdgcn_wmma_f32_16x16x4_f32(false, a[s], false, b0,
                                                 (short)0, c0, false, false);
      c1 = __builtin_amdgcn_wmma_f32_16x16x4_f32(false, a[s], false, b1,
                                                 (short)0, c1, false, false);
    }

    // score = |e|^2 - 2*sim  (row-constant |x|^2 is irrelevant to argmin)
    const int ci0 = t * TILEW + m;
    const int ci1 = t * TILEW + 16 + m;
#pragma unroll
    for (int j = 0; j < 8; ++j) {
      float s0 = en0 - 2.0f * c0[j];
      float s1 = en1 - 2.0f * c1[j];
      if (s0 < best[j]) { best[j] = s0; bidx[j] = ci0; }
      if (s1 < best[j]) { best[j] = s1; bidx[j] = ci1; }
    }
  }

  // Cross-lane argmin within each 16-lane half (rows j for lanes 0-15,
  // rows j+8 for lanes 16-31). xor masks 1,2,4,8 stay within each half.
#pragma unroll
  for (int j = 0; j < 8; ++j) {
#pragma unroll
    for (int off = 1; off < 16; off <<= 1) {
      float ob = __shfl_xor(best[j], off, 32);
      int   oi = __shfl_xor(bidx[j], off, 32);
      if (ob < best[j] || (ob == best[j] && oi < bidx[j])) {
        best[j] = ob; bidx[j] = oi;
      }
    }
  }
  if (m == 0) {
    int rbase = row0 + (lane >> 4) * 8;
#pragma unroll
    for (int j = 0; j < 8; ++j) idxOut[rbase + j] = bidx[j];
  }
}

// ---------------------------------------------------------------------------
// Kernel C: gather quantized vectors + loss = 1.25 * mean((q - x)^2)
// one block per row; 256 threads = D. Reads the transposed table (coalesced).
// ---------------------------------------------------------------------------
__global__ void vq_gather_loss_kernel(const float* __restrict__ x,
                                      const float* __restrict__ embT,
                                      const int* __restrict__ idx,
                                      float* __restrict__ out,
                                      float* __restrict__ loss) {
  const int n = blockIdx.x;
  const int d = threadIdx.x;
  const int k = idx[n];
  float q  = embT[(size_t)k * DDIM + d];
  float xv = x[(size_t)n * DDIM + d];
  out[(size_t)n * DDIM + d] = q;
  float v = (q - xv) * (q - xv);
#pragma unroll
  for (int off = 16; off > 0; off >>= 1) v += __shfl_down(v, off, 32);
  __shared__ float red[8];
  if ((threadIdx.x & 31) == 0) red[threadIdx.x >> 5] = v;
  __syncthreads();
  if (threadIdx.x == 0) {
    float s = 0.f;
#pragma unroll
    for (int i = 0; i < 8; ++i) s += red[i];
    atomicAdd(loss, s * (1.25f / (float)(N_ROWS * DDIM)));
  }
}

// ---------------------------------------------------------------------------
extern "C" void kernel_launch(void* const* d_in, const int* in_sizes, int n_in,
                              void* d_out, int out_size, void* d_ws,
                              size_t ws_size, hipStream_t stream) {
  const float* x   = (const float*)d_in[0];   // [32,32,32,256]
  const float* emb = (const float*)d_in[1];   // [256, 8192]
  float* out  = (float*)d_out;                // [N*256] quantized_ste
  float* loss = out + (out_size - 1);         // scalar loss (last element)

  float* enorm = (float*)d_ws;                                   // 32 KB
  int*   idx   = (int*)((char*)d_ws + KCODES * sizeof(float));   // 128 KB
  float* embT  = (float*)((char*)d_ws + KCODES * sizeof(float)
                          + N_ROWS * sizeof(int));               // 8 MB

  vq_transpose_kernel<<<dim3(KCODES / 32, DDIM / 32), 256, 0, stream>>>(emb, embT);
  vq_enorm_kernel<<<KCODES / 256, 256, 0, stream>>>(emb, enorm);
  vq_init_loss_kernel<<<1, 1, 0, stream>>>(loss);
  vq_argmin_kernel<<<N_ROWS / ROWS_PER_WG, 256, 0, stream>>>(x, embT, enorm, idx);
  vq_gather_loss_kernel<<<N_ROWS, 256, 0, stream>>>(x, embT, idx, out, loss);
}